// ContextQueryAttentionLayer_78847009620100
// MI455X (gfx1250) — compile-verified
//
#include <hip/hip_runtime.h>
#include <hip/hip_bf16.h>

// ---------------------------------------------------------------------------
// Context-Query attention (BiDAF) for MI455X / gfx1250, wave32 + WMMA f32.
// B=64, LC=1024, LQ=128, D=128.  All matrix contractions use
// V_WMMA_F32_16X16X4_F32.  LDS operands that are striped along K are stored
// pair-interleaved (float2 per {k, k+1}) so every WMMA fragment is a single
// conflict-free ds_load_b64 (kp-row stride == 32 mod 64 banks).
// Workspace layout (f32):
//   S      : B*LC*LQ        = 8388608
//   cw1    : B*LC           =   65536
//   qw2    : B*LQ           =    8192
//   rowmax : B*LC           =   65536
//   rowsum : B*LC           =   65536
//   colsum : B*LQ           =    8192
//   Tp     : B*LQ*D         = 1048576   (unnormalized T' in [j][d] layout)
//   total ~ 9.65M floats ~ 38.6 MB
// ---------------------------------------------------------------------------

typedef __attribute__((ext_vector_type(2))) float v2f;
typedef __attribute__((ext_vector_type(8))) float v8f;

#define BB   64
#define LCC  1024
#define LQQ  128
#define DD   128

// pair-interleaved B-operand strides (floats per kp-row); % 64 == 32
#define BSTR 288   // 128 columns * 2 + 32 pad
#define PSTR 96    // 32 columns * 2 + 32 pad

static __device__ __forceinline__ v8f wmma4(v2f a, v2f b, v8f c) {
  // D = A(16x4 f32) * B(4x16 f32) + C(16x16 f32)
  return __builtin_amdgcn_wmma_f32_16x16x4_f32(false, a, false, b, (short)0, c,
                                               false, false);
}

// ---------------------------------------------------------------------------
// K0: cw1[b,i] = c_rep[b,i,:].w1 ; qw2[b,j] = q_rep[b,j,:].w2
// One wave32 per row, 4 elements/lane, shuffle-xor reduce.
// ---------------------------------------------------------------------------
__global__ void __launch_bounds__(256)
k0_bias(const float* __restrict__ c_rep, const float* __restrict__ q_rep,
        const float* __restrict__ w, float* __restrict__ cw1,
        float* __restrict__ qw2) {
  const int gw   = (blockIdx.x * 256 + threadIdx.x) >> 5;
  const int lane = threadIdx.x & 31;
  const int nC = BB * LCC;
  const int nQ = BB * LQQ;
  const float* src;
  const float* wv;
  float* dst;
  if (gw < nC) {
    src = c_rep + (size_t)gw * DD; wv = w;       dst = cw1 + gw;
  } else {
    int r = gw - nC;
    if (r >= nQ) return;
    src = q_rep + (size_t)r * DD;  wv = w + DD;  dst = qw2 + r;
  }
  float s = 0.f;
  #pragma unroll
  for (int k = 0; k < DD; k += 32) s += src[k + lane] * wv[k + lane];
  #pragma unroll
  for (int off = 16; off > 0; off >>= 1) s += __shfl_xor(s, off, 32);
  if (lane == 0) *dst = s;
}

// ---------------------------------------------------------------------------
// K1: S = (c .* w3) @ q^T + cw1[i] + qw2[j]   (128x128 tile per workgroup)
//     + row-softmax stats (q_mask) -> rowmax, rowsum.  8 waves, 16 rows/wave.
// ---------------------------------------------------------------------------
__global__ void __launch_bounds__(256)
k1_scores(const float* __restrict__ c_rep, const float* __restrict__ q_rep,
          const int* __restrict__ q_mask, const float* __restrict__ w,
          const float* __restrict__ cw1, const float* __restrict__ qw2,
          float* __restrict__ S, float* __restrict__ rowmax,
          float* __restrict__ rowsum) {
  const int b   = blockIdx.y;
  const int i0  = blockIdx.x * 128;
  const int tid = threadIdx.x;
  const int lane = tid & 31;
  const int wv   = tid >> 5;
  const int nlo  = lane & 15;
  const int hi   = lane >> 4;

  extern __shared__ float sm[];
  float* c3   = sm;                 // [128][132]    c .* w3  (A, row-major)
  float* qTi  = c3 + 128 * 132;     // [64][BSTR]    q^T pair-interleaved on d
  float* cw1s = qTi + 64 * BSTR;    // [128]
  float* qw2s = cw1s + 128;         // [128]
  float* qms  = qw2s + 128;         // [128]

  const float* cb = c_rep + ((size_t)b * LCC + i0) * DD;
  const float* qb = q_rep + (size_t)b * LQQ * DD;
  for (int idx = tid; idx < 128 * 128; idx += 256) {
    int r = idx >> 7, d = idx & 127;                 // r = j (or c-row), d
    c3[r * 132 + d] = cb[idx] * w[256 + d];
    qTi[(d >> 1) * BSTR + 2 * r + (d & 1)] = qb[idx]; // pair {d, d+1} at col j=r
  }
  if (tid < 128) {
    cw1s[tid] = cw1[b * LCC + i0 + tid];
    qw2s[tid] = qw2[b * LQQ + tid];
    qms[tid]  = (q_mask[b * LQQ + tid] != 0) ? 1.0f : 0.0f;
  }
  __syncthreads();

  const v8f vz = {0.f, 0.f, 0.f, 0.f, 0.f, 0.f, 0.f, 0.f};
  v8f acc[8];
  #pragma unroll
  for (int t = 0; t < 8; ++t) acc[t] = vz;

  const int m = wv * 16 + nlo;            // A-matrix row (output row in strip)
  #pragma unroll 4
  for (int kk = 0; kk < DD; kk += 4) {
    const int kp = (kk >> 1) + hi;        // k0 = kk + 2*hi
    v2f a = *(const v2f*)&c3[m * 132 + kk + 2 * hi];
    const float* brow = qTi + kp * BSTR;
    #pragma unroll
    for (int t = 0; t < 8; ++t) {
      v2f bf = *(const v2f*)&brow[2 * (t * 16 + nlo)];
      acc[t] = wmma4(a, bf, acc[t]);
    }
  }

  // add cw1[i] + qw2[j]
  #pragma unroll
  for (int t = 0; t < 8; ++t) {
    const float qw = qw2s[t * 16 + nlo];
    #pragma unroll
    for (int r = 0; r < 8; ++r)
      acc[t][r] += cw1s[wv * 16 + r + 8 * hi] + qw;
  }

  // row softmax stats (mask by q_mask) + store raw S
  #pragma unroll
  for (int r = 0; r < 8; ++r) {
    float mx = -3.0e38f;
    #pragma unroll
    for (int t = 0; t < 8; ++t) {
      float v = (qms[t * 16 + nlo] != 0.f) ? -1e30f : acc[t][r];
      mx = fmaxf(mx, v);
    }
    #pragma unroll
    for (int off = 1; off < 16; off <<= 1) mx = fmaxf(mx, __shfl_xor(mx, off, 32));
    float se = 0.f;
    #pragma unroll
    for (int t = 0; t < 8; ++t) {
      float v = (qms[t * 16 + nlo] != 0.f) ? -1e30f : acc[t][r];
      se += __expf(v - mx);
    }
    #pragma unroll
    for (int off = 1; off < 16; off <<= 1) se += __shfl_xor(se, off, 32);

    const int grow = i0 + wv * 16 + r + 8 * hi;
    if (nlo == 0) {
      rowmax[b * LCC + grow] = mx;
      rowsum[b * LCC + grow] = se;
    }
    float* Srow = S + ((size_t)b * LCC + grow) * LQQ;
    #pragma unroll
    for (int t = 0; t < 8; ++t) Srow[t * 16 + nlo] = acc[t][r];
  }
}

// ---------------------------------------------------------------------------
// K2: column softmax (c_mask) over i for a 32-column chunk:
//     colmax pass, then P = exp(S_c - colmax), colsum += P,
//     T'^T[d][j] += c^T @ P via WMMA.  T' stored as Tp[b][j][d].
// Both operands pair-interleaved along i (the K dim).
// ---------------------------------------------------------------------------
__global__ void __launch_bounds__(256)
k2_colpass(const float* __restrict__ c_rep, const int* __restrict__ c_mask,
           const float* __restrict__ S, float* __restrict__ colsum,
           float* __restrict__ Tp) {
  const int b   = blockIdx.y;
  const int j0  = blockIdx.x * 32;
  const int tid = threadIdx.x;
  const int lane = tid & 31;
  const int wv   = tid >> 5;
  const int nlo  = lane & 15;
  const int hi   = lane >> 4;
  const int jl   = tid & 31;

  extern __shared__ float sm[];
  float* cbi = sm;                  // [16][BSTR] c block pair-interleaved on i
  float* pbi = cbi + 16 * BSTR;     // [16][PSTR] P block pair-interleaved on i
  float* cmx = pbi + 16 * PSTR;     // [32] column max
  float* red = cmx + 32;            // [256] reduce scratch

  const float* Sb  = S + (size_t)b * LCC * LQQ;
  const int*   cmk = c_mask + b * LCC;

  // pass 1: column max over all i
  float lm = -3.0e38f;
  for (int ib = 0; ib < LCC; ib += 32) {
    #pragma unroll
    for (int s = 0; s < 4; ++s) {
      const int i = ib + (tid >> 5) + 8 * s;
      float v = Sb[(size_t)i * LQQ + j0 + jl];
      lm = fmaxf(lm, (cmk[i] != 0) ? -1e30f : v);
    }
  }
  red[tid] = lm;
  __syncthreads();
  if (tid < 32) {
    float mv = red[tid];
    #pragma unroll
    for (int k = 1; k < 8; ++k) mv = fmaxf(mv, red[tid + 32 * k]);
    cmx[tid] = mv;
  }
  __syncthreads();

  const v8f vz = {0.f, 0.f, 0.f, 0.f, 0.f, 0.f, 0.f, 0.f};
  v8f acc0 = vz, acc1 = vz;         // T'^T tiles: d-tile = wave, j-tiles 0/1
  float csl = 0.f;
  const int dt = wv;

  for (int ib = 0; ib < LCC; ib += 32) {
    // prefetch next c block (global_prefetch_b8)
    if (ib + 32 < LCC)
      __builtin_prefetch(c_rep + ((size_t)b * LCC + ib + 32) * DD + tid * 16, 0, 0);

    const float* cbk = c_rep + ((size_t)b * LCC + ib) * DD;
    for (int idx = tid; idx < 32 * 128; idx += 256) {
      int r = idx >> 7, d = idx & 127;
      cbi[(r >> 1) * BSTR + 2 * d + (r & 1)] = cbk[idx];
    }
    #pragma unroll
    for (int s = 0; s < 4; ++s) {
      const int il = (tid >> 5) + 8 * s;
      const int i  = ib + il;
      float v = Sb[(size_t)i * LQQ + j0 + jl];
      float p = __expf(((cmk[i] != 0) ? -1e30f : v) - cmx[jl]);
      pbi[(il >> 1) * PSTR + 2 * jl + (il & 1)] = p;
      csl += p;
    }
    __syncthreads();
    #pragma unroll
    for (int kk = 0; kk < 32; kk += 4) {
      const int kp = (kk >> 1) + hi;
      v2f a  = *(const v2f*)&cbi[kp * BSTR + 2 * (dt * 16 + nlo)];
      v2f b0 = *(const v2f*)&pbi[kp * PSTR + 2 * nlo];
      v2f b1 = *(const v2f*)&pbi[kp * PSTR + 2 * (16 + nlo)];
      acc0 = wmma4(a, b0, acc0);
      acc1 = wmma4(a, b1, acc1);
    }
    __syncthreads();
  }

  red[tid] = csl;
  __syncthreads();
  if (tid < 32) {
    float sv = red[tid];
    #pragma unroll
    for (int k = 1; k < 8; ++k) sv += red[tid + 32 * k];
    colsum[b * LQQ + j0 + tid] = sv;
  }

  float* Tb = Tp + (size_t)b * LQQ * DD;
  #pragma unroll
  for (int r = 0; r < 8; ++r) {
    const int d = dt * 16 + r + 8 * hi;
    Tb[(size_t)(j0 + nlo) * DD + d]      = acc0[r];
    Tb[(size_t)(j0 + 16 + nlo) * DD + d] = acc1[r];
  }
}

// ---------------------------------------------------------------------------
// K4: S1 tile from raw S + row stats; A = S1@q ; B_att = S1@(T'/colsum);
//     out = [c, A, c.*A, c.*B_att]  (128 rows per workgroup).
// B operand pair-interleaved along j (the K dim).
// ---------------------------------------------------------------------------
__global__ void __launch_bounds__(256)
k4_output(const float* __restrict__ c_rep, const float* __restrict__ q_rep,
          const int* __restrict__ q_mask, const float* __restrict__ S,
          const float* __restrict__ rowmax, const float* __restrict__ rowsum,
          const float* __restrict__ colsum, const float* __restrict__ Tp,
          float* __restrict__ out) {
  const int b   = blockIdx.y;
  const int i0  = blockIdx.x * 128;
  const int tid = threadIdx.x;
  const int lane = tid & 31;
  const int wv   = tid >> 5;
  const int nlo  = lane & 15;
  const int hi   = lane >> 4;

  extern __shared__ float sm[];
  float* p1  = sm;                  // [128][132] S1 tile (A, row-major)
  float* bbf = p1 + 128 * 132;      // [64][BSTR] B operand pair-interleaved on j
  float* rmx = bbf + 64 * BSTR;     // [128]
  float* rin = rmx + 128;           // [128] 1/rowsum
  float* qms = rin + 128;           // [128]
  float* cin = qms + 128;           // [128] 1/colsum

  if (tid < 128) {
    rmx[tid] = rowmax[b * LCC + i0 + tid];
    rin[tid] = 1.0f / rowsum[b * LCC + i0 + tid];
    qms[tid] = (q_mask[b * LQQ + tid] != 0) ? 1.0f : 0.0f;
    cin[tid] = 1.0f / colsum[b * LQQ + tid];
  }
  __syncthreads();

  const float* Sb = S + ((size_t)b * LCC + i0) * LQQ;
  const float* qb = q_rep + (size_t)b * LQQ * DD;
  for (int idx = tid; idx < 128 * 128; idx += 256) {
    int r = idx >> 7, cc = idx & 127;                 // r = row i (or j), cc = d
    float v = (qms[cc] != 0.f) ? -1e30f : Sb[idx];
    p1[r * 132 + cc] = __expf(v - rmx[r]) * rin[r];
    bbf[(r >> 1) * BSTR + 2 * cc + (r & 1)] = qb[idx]; // pair {j, j+1} at col d
  }
  __syncthreads();

  const v8f vz = {0.f, 0.f, 0.f, 0.f, 0.f, 0.f, 0.f, 0.f};
  const int m = wv * 16 + nlo;
  const float* crow = c_rep + ((size_t)b * LCC + i0) * DD;

  // ---- GEMM1: A = S1 @ q ----
  v8f acc[8];
  #pragma unroll
  for (int t = 0; t < 8; ++t) acc[t] = vz;
  #pragma unroll 4
  for (int kk = 0; kk < LQQ; kk += 4) {
    const int kp = (kk >> 1) + hi;
    v2f a = *(const v2f*)&p1[m * 132 + kk + 2 * hi];
    const float* brow = bbf + kp * BSTR;
    #pragma unroll
    for (int t = 0; t < 8; ++t) {
      v2f bf = *(const v2f*)&brow[2 * (t * 16 + nlo)];
      acc[t] = wmma4(a, bf, acc[t]);
    }
  }
  #pragma unroll
  for (int nt = 0; nt < 8; ++nt) {
    #pragma unroll
    for (int r = 0; r < 8; ++r) {
      const int grow = wv * 16 + r + 8 * hi;
      const int d    = nt * 16 + nlo;
      const float cv = crow[(size_t)grow * DD + d];
      const float av = acc[nt][r];
      float* op = out + ((size_t)b * LCC + i0 + grow) * (4 * DD);
      op[d]          = cv;
      op[DD + d]     = av;
      op[2 * DD + d] = cv * av;
    }
  }
  __syncthreads();

  // restage B = T = T' / colsum  (pair-interleaved on j)
  const float* Tb = Tp + (size_t)b * LQQ * DD;
  for (int idx = tid; idx < 128 * 128; idx += 256) {
    int j = idx >> 7, d = idx & 127;
    bbf[(j >> 1) * BSTR + 2 * d + (j & 1)] = Tb[idx] * cin[j];
  }
  __syncthreads();

  // ---- GEMM2: B_att = S1 @ T ----
  #pragma unroll
  for (int t = 0; t < 8; ++t) acc[t] = vz;
  #pragma unroll 4
  for (int kk = 0; kk < LQQ; kk += 4) {
    const int kp = (kk >> 1) + hi;
    v2f a = *(const v2f*)&p1[m * 132 + kk + 2 * hi];
    const float* brow = bbf + kp * BSTR;
    #pragma unroll
    for (int t = 0; t < 8; ++t) {
      v2f bf = *(const v2f*)&brow[2 * (t * 16 + nlo)];
      acc[t] = wmma4(a, bf, acc[t]);
    }
  }
  #pragma unroll
  for (int nt = 0; nt < 8; ++nt) {
    #pragma unroll
    for (int r = 0; r < 8; ++r) {
      const int grow = wv * 16 + r + 8 * hi;
      const int d    = nt * 16 + nlo;
      const float cv = crow[(size_t)grow * DD + d];
      float* op = out + ((size_t)b * LCC + i0 + grow) * (4 * DD);
      op[3 * DD + d] = cv * acc[nt][r];
    }
  }
}

// ---------------------------------------------------------------------------
extern "C" void kernel_launch(void* const* d_in, const int* in_sizes, int n_in,
                              void* d_out, int out_size, void* d_ws,
                              size_t ws_size, hipStream_t stream) {
  (void)in_sizes; (void)n_in; (void)out_size; (void)ws_size;
  const float* c_rep  = (const float*)d_in[0];
  const float* q_rep  = (const float*)d_in[1];
  const int*   c_mask = (const int*)d_in[2];
  const int*   q_mask = (const int*)d_in[3];
  const float* w      = (const float*)d_in[4];
  float* out = (float*)d_out;

  float* ws     = (float*)d_ws;
  float* S      = ws;                      // 8388608
  float* cw1    = S + (size_t)BB * LCC * LQQ;
  float* qw2    = cw1 + BB * LCC;
  float* rowmax = qw2 + BB * LQQ;
  float* rowsum = rowmax + BB * LCC;
  float* colsum = rowsum + BB * LCC;
  float* Tp     = colsum + BB * LQQ;

  const size_t smem1 = (size_t)(128 * 132 + 64 * BSTR + 3 * 128) * 4;
  const size_t smem2 = (size_t)(16 * BSTR + 16 * PSTR + 32 + 256) * 4;
  const size_t smem4 = (size_t)(128 * 132 + 64 * BSTR + 4 * 128) * 4;

  // K0: (B*LC + B*LQ) rows, 1 wave per row, 8 waves/block
  const int rows = BB * LCC + BB * LQQ;
  k0_bias<<<rows / 8, 256, 0, stream>>>(c_rep, q_rep, w, cw1, qw2);

  k1_scores<<<dim3(LCC / 128, BB), 256, smem1, stream>>>(
      c_rep, q_rep, q_mask, w, cw1, qw2, S, rowmax, rowsum);

  k2_colpass<<<dim3(LQQ / 32, BB), 256, smem2, stream>>>(
      c_rep, c_mask, S, colsum, Tp);

  k4_output<<<dim3(LCC / 128, BB), 256, smem4, stream>>>(
      c_rep, q_rep, q_mask, S, rowmax, rowsum, colsum, Tp, out);
}